// FCOSSparsePostProcessor_30408368456271
// MI455X (gfx1250) — compile-verified
//
#include <hip/hip_runtime.h>

#define M_PTS   2097152
#define NCLS    11
#define NIMG    16
#define NBINS   4096
#define CAP     4096
#define KSLOT   1024
#define PRE_N   1000
#define POST_N  100
#define KW      32          // 1024 bits -> 32 words
#define HGRID   512         // histogram blocks
#define PPB     (M_PTS / HGRID)

typedef __attribute__((ext_vector_type(2))) float v2f;
typedef __attribute__((ext_vector_type(8))) float v8f;

static __device__ __forceinline__ float fast_rcp(float x) {
#if __has_builtin(__builtin_amdgcn_rcpf)
    return __builtin_amdgcn_rcpf(x);      // v_rcp_f32, 1 ulp
#else
    return 1.0f / x;
#endif
}

static __device__ __forceinline__ float sigm(float x) {
    return fast_rcp(1.0f + __expf(-x));
}

static __device__ __forceinline__ unsigned ballot32(bool p) {
#if __has_builtin(__builtin_amdgcn_ballot_w32)
    return __builtin_amdgcn_ballot_w32(p);
#else
    return (unsigned)__ballot(p);
#endif
}

// bit-exact monotonic binning of score in (0.05, 1.0]
#define LO_BITS 0x3D4CCCCDu   // bits(0.05f)
#define HI_BITS 0x3F800000u   // bits(1.0f)
static __device__ __forceinline__ int score_bin(float s) {
    unsigned u = __float_as_uint(s);
    unsigned long long num = (unsigned long long)(u - LO_BITS) * (unsigned long long)NBINS;
    int b = (int)(num / (unsigned long long)(HI_BITS - LO_BITS));
    return b >= NBINS ? (NBINS - 1) : b;
}

// ---------------------------------------------------------------- init
__global__ void k_init(unsigned* __restrict__ hist, unsigned* __restrict__ cnt) {
    int i = blockIdx.x * blockDim.x + threadIdx.x;
    int total = NIMG * NBINS;
    for (; i < total + NIMG; i += gridDim.x * blockDim.x) {
        if (i < total) hist[i] = 0u;
        else cnt[i - total] = 0u;
    }
}

// ---------------------------------------------------------------- pass 1: histogram
__global__ void __launch_bounds__(256) k_hist(const float* __restrict__ cls,
                                              const float* __restrict__ cent,
                                              const int*   __restrict__ batch,
                                              unsigned*    __restrict__ hist) {
    __shared__ unsigned lh[2][NBINS];     // 32KB: two image-private histograms
    __shared__ int simg[2];
    const int start = blockIdx.x * PPB;
    if (threadIdx.x == 0) { simg[0] = batch[start]; simg[1] = batch[start + PPB - 1]; }
    for (int i = threadIdx.x; i < 2 * NBINS; i += blockDim.x) (&lh[0][0])[i] = 0u;
    __syncthreads();
    const int i0 = simg[0], i1 = simg[1];

    for (int p = start + threadIdx.x; p < start + PPB; p += blockDim.x) {
        const int img = batch[p];
        const float sc_c = sigm(cent[p]);
        const float* row = cls + (size_t)p * NCLS;
        __builtin_prefetch(row + (size_t)blockDim.x * NCLS, 0, 1);  // global_prefetch_b8
#pragma unroll
        for (int c = 0; c < NCLS - 1; ++c) {          // last class zeroed in reference
            float s = sigm(row[c]) * sc_c;
            if (s > 0.05f) {
                int b = score_bin(s);
                if (img == i0)      atomicAdd(&lh[0][b], 1u);
                else if (img == i1) atomicAdd(&lh[1][b], 1u);
                else                atomicAdd(&hist[img * NBINS + b], 1u); // rare
            }
        }
    }
    __syncthreads();
    for (int b = threadIdx.x; b < NBINS; b += blockDim.x) {
        unsigned v0 = lh[0][b];
        if (v0) atomicAdd(&hist[i0 * NBINS + b], v0);
        if (i1 != i0) {
            unsigned v1 = lh[1][b];
            if (v1) atomicAdd(&hist[i1 * NBINS + b], v1);
        }
    }
}

// ---------------------------------------------------------------- cutoff-bin selection
__global__ void __launch_bounds__(512) k_select(const unsigned* __restrict__ hist,
                                                unsigned* __restrict__ cutoff) {
    const int w = threadIdx.x >> 5;       // wave = image
    const int lane = threadIdx.x & 31;
    if (w >= NIMG) return;
    const unsigned* h = hist + w * NBINS;
    unsigned run = 0;
    int cut = 0;
    for (int base = NBINS - 32; base >= 0; base -= 32) {
        unsigned v = h[base + lane];
        unsigned s = v;                                      // suffix-inclusive scan
#pragma unroll
        for (int off = 1; off < 32; off <<= 1) {
            unsigned t = __shfl_down(s, off, 32);
            if (lane + off < 32) s += t;
        }
        unsigned total = __shfl(s, 0, 32);
        if (run + total >= PRE_N) {                          // uniform branch
            unsigned bal = ballot32((run + s) >= PRE_N);     // true on a lane prefix
            int L = 31 - __clz(bal);
            cut = base + L;
            break;
        }
        run += total;
    }
    if (lane == 0) cutoff[w] = (unsigned)cut;
}

// ---------------------------------------------------------------- pass 2: compaction
__global__ void __launch_bounds__(256) k_compact(const float* __restrict__ cls,
                                                 const float* __restrict__ cent,
                                                 const int*   __restrict__ batch,
                                                 const unsigned* __restrict__ cutoff,
                                                 unsigned* __restrict__ cnt,
                                                 unsigned long long* __restrict__ cand) {
    const int p = blockIdx.x * blockDim.x + threadIdx.x;
    if (p >= M_PTS) return;
    const int img = batch[p];
    const unsigned cimg = cutoff[img];
    const float sc_c = sigm(cent[p]);
    const float* row = cls + (size_t)p * NCLS;
    __builtin_prefetch(row + (size_t)blockDim.x * NCLS, 0, 1);
#pragma unroll
    for (int c = 0; c < NCLS - 1; ++c) {
        float s = sigm(row[c]) * sc_c;
        if (s > 0.05f) {
            int b = score_bin(s);
            if ((unsigned)b >= cimg) {
                unsigned pos = atomicAdd(&cnt[img], 1u);
                if (pos < CAP) {
                    unsigned flat = (unsigned)p * NCLS + (unsigned)c;
                    cand[(size_t)img * CAP + pos] =
                        ((unsigned long long)__float_as_uint(s) << 32) | (unsigned)(~flat);
                }
            }
        }
    }
}

// ---------------------------------------------------------------- per-image NMS + output
__global__ void __launch_bounds__(512) k_nms(const unsigned long long* __restrict__ cand,
                                             const unsigned* __restrict__ cnt,
                                             const float* __restrict__ locations,
                                             const int*   __restrict__ levels,
                                             const float* __restrict__ breg,
                                             const int*   __restrict__ isz,
                                             float* __restrict__ out) {
    __shared__ union {
        unsigned long long keys[CAP];     // 32KB  (phase A only)
        unsigned sup[KSLOT * KW];         // 128KB (suppression bit-matrix)
    } u;
    __shared__ unsigned selFlat[KSLOT];   // 4KB
    __shared__ float    sc[KSLOT];        // 4KB  topv -> sqrt -> kept score
    __shared__ float4   boff[KSLOT];      // 16KB offset boxes
    __shared__ float    offv[KSLOT];      // 4KB  class -> class*(maxc+1)
    __shared__ unsigned keepw[KW];
    __shared__ float    red[16];
    __shared__ float    s_maxc;

    const int img  = blockIdx.x;
    const int tid  = threadIdx.x;
    const int nt   = blockDim.x;
    const int wid  = tid >> 5, lane = tid & 31, nwave = nt >> 5;

    int n = (int)cnt[img]; if (n > CAP) n = CAP;

    // ---- A: exact top-PRE_N via rank selection over <=CAP keys
    for (int i = tid; i < n; i += nt) u.keys[i] = cand[(size_t)img * CAP + i];
    for (int i = tid; i < KSLOT; i += nt) { selFlat[i] = 0xFFFFFFFFu; sc[i] = 0.0f; }
    __syncthreads();
    for (int i = tid; i < n; i += nt) {
        unsigned long long ki = u.keys[i];
        int r = 0;
        for (int j = 0; j < n; ++j) r += (u.keys[j] > ki);
        if (r < PRE_N) {
            selFlat[r] = ~((unsigned)ki);
            sc[r] = __uint_as_float((unsigned)(ki >> 32));    // topv
        }
    }
    __syncthreads();

    // ---- B: decode, clip, score, max_coord reduce, class offset
    const int hh = isz[2 * img + 0], ww = isz[2 * img + 1];
    const float h = (float)hh, w = (float)ww;
    float maxc_local = 0.0f;
    for (int s = tid; s < KSLOT; s += nt) {
        unsigned flat = selFlat[s];
        float topv = sc[s];
        float4 b4 = make_float4(0.f, 0.f, 0.f, 0.f);
        float clv = 0.0f, scv = 0.0f;
        if (topv > 0.0f && flat != 0xFFFFFFFFu) {
            unsigned pt = flat / NCLS;
            unsigned c0 = flat - pt * NCLS;
            clv = (float)(c0 + 1);
            float lx = locations[2 * pt], ly = locations[2 * pt + 1];
            float stride = (float)(8 << levels[pt]);
            float r0 = breg[4 * pt + 0] * stride, r1 = breg[4 * pt + 1] * stride;
            float r2 = breg[4 * pt + 2] * stride, r3 = breg[4 * pt + 3] * stride;
            float x1 = fminf(fmaxf(lx - r0, 0.f), w - 1.0f);
            float y1 = fminf(fmaxf(ly - r1, 0.f), h - 1.0f);
            float x2 = fminf(fmaxf(lx + r2, 0.f), w - 1.0f);
            float y2 = fminf(fmaxf(ly + r3, 0.f), h - 1.0f);
            b4 = make_float4(x1, y1, x2, y2);
            scv = sqrtf(topv);
            maxc_local = fmaxf(maxc_local, fmaxf(fmaxf(x1, y1), fmaxf(x2, y2)));
        }
        sc[s] = scv;
        boff[s] = b4;       // un-offset for now
        offv[s] = clv;      // class (0 if invalid)
    }
#pragma unroll
    for (int o = 16; o > 0; o >>= 1)
        maxc_local = fmaxf(maxc_local, __shfl_down(maxc_local, o, 32));
    if (lane == 0) red[wid] = maxc_local;
    __syncthreads();
    if (tid == 0) {
        float m = 0.0f;
        for (int i = 0; i < nwave; ++i) m = fmaxf(m, red[i]);
        s_maxc = m;
    }
    __syncthreads();
    const float off1 = s_maxc + 1.0f;
    for (int s = tid; s < KSLOT; s += nt) {
        float o = offv[s] * off1;
        offv[s] = o;
        float4 b4 = boff[s];
        b4.x += o; b4.y += o; b4.z += o; b4.w += o;
        boff[s] = b4;
    }
    __syncthreads();

    // ---- C: suppression bit-matrix via per-wave 16x16 WMMA tiles
    for (int i = tid; i < KSLOT * KW; i += nt) u.sup[i] = 0u;
    __syncthreads();

    const int TPS = KSLOT / 16;                    // 64 tiles per side
    for (int t = wid; t < TPS * TPS; t += nwave) {
        const int m0 = (t / TPS) * 16, n0 = (t % TPS) * 16;
        const bool lo = lane < 16;
        const int  li = lane & 15;
        const float4 bn  = boff[n0 + li];
        const float4 bma = boff[m0 + li];
        const float areaN = (bn.z - bn.x + 1.0f) * (bn.w - bn.y + 1.0f);
        const float areaM = (bma.z - bma.x + 1.0f) * (bma.w - bma.y + 1.0f);
#if __has_builtin(__builtin_amdgcn_wmma_f32_16x16x4_f32)
        // D[m][n] = area_m * 1 + 1 * area_n  (outer sum = union base)
        v2f A, B;
        A[0] = lo ? areaM : 0.0f;  A[1] = lo ? 1.0f  : 0.0f;   // lanes16-31: K=2,3 -> 0
        B[0] = lo ? 1.0f  : 0.0f;  B[1] = lo ? areaN : 0.0f;
        v8f Cz = {0.f,0.f,0.f,0.f,0.f,0.f,0.f,0.f};
        v8f D = __builtin_amdgcn_wmma_f32_16x16x4_f32(false, A, false, B, (short)0, Cz,
                                                      false, false);
#endif
        const int mhi = lo ? 0 : 8;
#pragma unroll
        for (int k = 0; k < 8; ++k) {
            const int m = m0 + k + mhi;            // matches C/D VGPR layout
            float4 bm = boff[m];
            float xx1 = fmaxf(bm.x, bn.x), yy1 = fmaxf(bm.y, bn.y);
            float xx2 = fminf(bm.z, bn.z), yy2 = fminf(bm.w, bn.w);
            float inter = fmaxf(xx2 - xx1 + 1.0f, 0.0f) * fmaxf(yy2 - yy1 + 1.0f, 0.0f);
#if __has_builtin(__builtin_amdgcn_wmma_f32_16x16x4_f32)
            float asum = D[k];
#else
            float asum = (bm.z - bm.x + 1.0f) * (bm.w - bm.y + 1.0f) + areaN;
#endif
            // iou > 0.6  <=>  inter > 0.6 * union   (union > 0): no division needed
            float uni = fmaxf(asum - inter, 1e-9f);
            unsigned bal = ballot32(inter > 0.6f * uni);
            if (lane == 0)
                atomicOr(&u.sup[(m0 + k) * KW + (n0 >> 5)], (bal & 0xFFFFu) << (n0 & 16));
            if (lane == 16)
                atomicOr(&u.sup[(m0 + k + 8) * KW + (n0 >> 5)], (bal >> 16) << (n0 & 16));
        }
    }
    __syncthreads();

    // ---- D: greedy scan, single wave owns the whole 1024-bit keep mask
    if (wid == 0) {
        unsigned kwv = 0u;
#pragma unroll
        for (int b = 0; b < 32; ++b)
            kwv |= (sc[lane * 32 + b] > 0.0f) ? (1u << b) : 0u;
        for (int i = 0; i < PRE_N; ++i) {
            const int iw_ = i >> 5, ib = i & 31;
            unsigned kword = __shfl(kwv, iw_, 32);
            if ((kword >> ib) & 1u) {              // uniform
                unsigned m = u.sup[i * KW + lane];
                if (lane < iw_)       m = 0u;
                else if (lane == iw_) m &= ~((2u << ib) - 1u);  // only j > i
                kwv &= ~m;
            }
        }
        keepw[lane] = kwv;
    }
    __syncthreads();

    // ---- E: exact top-POST_N of kept scores, write outputs
    for (int s = tid; s < KSLOT; s += nt) {
        bool kp = (keepw[s >> 5] >> (s & 31)) & 1u;
        sc[s] = kp ? sc[s] : 0.0f;
    }
    __syncthreads();
    for (int s = tid; s < KSLOT; s += nt) {
        unsigned long long ks =
            ((unsigned long long)__float_as_uint(sc[s]) << 32) | (unsigned)(KSLOT - s);
        int r = 0;
        for (int j = 0; j < KSLOT; ++j) {
            unsigned long long kj =
                ((unsigned long long)__float_as_uint(sc[j]) << 32) | (unsigned)(KSLOT - j);
            r += (kj > ks);
        }
        if (r < POST_N) {
            float o = offv[s];
            float4 b4 = boff[s];
            float* ob = out + ((size_t)img * POST_N + r) * 4;
            ob[0] = b4.x - o; ob[1] = b4.y - o; ob[2] = b4.z - o; ob[3] = b4.w - o;
            float fv = sc[s];
            unsigned flat = selFlat[s];
            out[NIMG * POST_N * 4 + img * POST_N + r] = fv;
            out[NIMG * POST_N * 5 + img * POST_N + r] =
                (flat != 0xFFFFFFFFu) ? (float)(flat % NCLS + 1u) : 0.0f;
            out[NIMG * POST_N * 6 + img * POST_N + r] = (fv > 0.0f) ? 1.0f : 0.0f;
        }
    }
}

// ---------------------------------------------------------------- host
extern "C" void kernel_launch(void* const* d_in, const int* in_sizes, int n_in,
                              void* d_out, int out_size, void* d_ws, size_t ws_size,
                              hipStream_t stream) {
    (void)in_sizes; (void)n_in; (void)out_size; (void)ws_size;
    const float* locations = (const float*)d_in[0];
    const int*   levels    = (const int*)d_in[1];
    const int*   batch     = (const int*)d_in[2];
    const float* box_cls   = (const float*)d_in[3];
    const float* breg      = (const float*)d_in[4];
    const float* cent      = (const float*)d_in[5];
    const int*   isz       = (const int*)d_in[6];
    float* out = (float*)d_out;

    char* ws = (char*)d_ws;
    unsigned* hist = (unsigned*)ws;                              // 16*4096*4 = 256KB
    unsigned* cnt  = (unsigned*)(ws + 262144);                   // 16
    unsigned* cut  = (unsigned*)(ws + 262144 + 64);              // 16
    unsigned long long* cand = (unsigned long long*)(ws + 262272); // 16*4096*8 = 512KB

    k_init   <<<64,   256, 0, stream>>>(hist, cnt);
    k_hist   <<<HGRID,256, 0, stream>>>(box_cls, cent, batch, hist);
    k_select <<<1,    512, 0, stream>>>(hist, cut);
    k_compact<<<M_PTS/256, 256, 0, stream>>>(box_cls, cent, batch, cut, cnt, cand);
    k_nms    <<<NIMG, 512, 0, stream>>>(cand, cnt, locations, levels, breg, isz, out);
}